// AdaptiveRouter_50534585205486
// MI455X (gfx1250) — compile-verified
//
#include <hip/hip_runtime.h>
#include <hip/hip_bf16.h>
#include <math.h>

typedef __attribute__((ext_vector_type(2))) float v2f;
typedef __attribute__((ext_vector_type(8))) float v8f;

#define RT_N 16384
#define RT_H 4096
#define RT_E 64
#define RT_EPS 1e-8f

// ---------------------------------------------------------------------------
// Kernel 0: bias = log(softmax(importance)+eps), inv_temp, zero accumulators.
// ws layout (floats): [0..63] bias, [64] inv_temp, [65..128] expert_load acc,
//                     [129] entropy acc, [256...] packed B fragments (1MB).
// ---------------------------------------------------------------------------
__global__ void router_prep_kernel(const float* __restrict__ importance,
                                   const float* __restrict__ temperature,
                                   float* __restrict__ ws) {
    __shared__ float sm[64];
    int t = threadIdx.x;              // 64 threads
    float v = importance[t];
    sm[t] = v;
    __syncthreads();
    float mx = sm[0];
    for (int i = 1; i < 64; ++i) mx = fmaxf(mx, sm[i]);
    float ex = __expf(v - mx);
    __syncthreads();
    sm[t] = ex;
    __syncthreads();
    float s = 0.f;
    for (int i = 0; i < 64; ++i) s += sm[i];
    ws[t] = __logf(ex / s + RT_EPS);      // bias[e]
    ws[65 + t] = 0.f;                     // zero expert_load accumulator
    if (t == 0) {
        ws[64]  = 1.0f / temperature[0];  // inv temperature
        ws[129] = 0.f;                    // zero entropy accumulator
    }
}

// ---------------------------------------------------------------------------
// Kernel 1: pack router_weight [E,H] into WMMA-B fragment order.
// Entry index = (k4*4 + nt)*32 + lane ; each entry = 2 floats (VGPR0,VGPR1).
// B[k][n] = W[n][k];  lane<16 -> K = 4*k4 + {0,1}, lane>=16 -> K = 4*k4+{2,3};
// n = nt*16 + (lane&15).
// ---------------------------------------------------------------------------
__global__ void router_pack_kernel(const float* __restrict__ w,
                                   float* __restrict__ bpack) {
    int idx = blockIdx.x * 256 + threadIdx.x;     // 131072 entries
    if (idx >= (RT_H / 4) * 4 * 32) return;
    int lane = idx & 31;
    int nt   = (idx >> 5) & 3;
    int k4   = idx >> 7;
    int e    = nt * 16 + (lane & 15);
    int k    = k4 * 4 + ((lane >> 4) << 1);
    const float* src = w + (size_t)e * RT_H + k;
    bpack[(size_t)idx * 2 + 0] = src[0];
    bpack[(size_t)idx * 2 + 1] = src[1];
}

// ---------------------------------------------------------------------------
// Kernel 2: WMMA GEMM. 8 waves/block, each wave -> 16 rows x 64 experts.
// A (16x4 f32): lane<16 holds row=lane, K={0,1}; lane>=16 row=lane-16, K={2,3}.
// ---------------------------------------------------------------------------
__global__ __launch_bounds__(256) void router_gemm_kernel(
    const float* __restrict__ hs,
    const float* __restrict__ bpack,
    const float* __restrict__ ws,
    float* __restrict__ logits) {
    const int lane = threadIdx.x & 31;
    const int wave = threadIdx.x >> 5;
    const int row_base = (blockIdx.x * 8 + wave) * 16;
    const int m   = lane & 15;
    const int klo = (lane >> 4) << 1;       // 0 for lanes 0-15, 2 for 16-31

    const float* arow = hs + (size_t)(row_base + m) * RT_H + klo;

    v8f acc[4] = {};

    #pragma unroll 4
    for (int k4 = 0; k4 < RT_H / 4; ++k4) {
        v2f a = *(const v2f*)(arow + k4 * 4);
        const float* bk = bpack + (size_t)k4 * 256 + (size_t)lane * 2;
        v2f b0 = *(const v2f*)(bk + 0 * 64);
        v2f b1 = *(const v2f*)(bk + 1 * 64);
        v2f b2 = *(const v2f*)(bk + 2 * 64);
        v2f b3 = *(const v2f*)(bk + 3 * 64);
        acc[0] = __builtin_amdgcn_wmma_f32_16x16x4_f32(false, a, false, b0,
                                                       (short)0, acc[0], false, false);
        acc[1] = __builtin_amdgcn_wmma_f32_16x16x4_f32(false, a, false, b1,
                                                       (short)0, acc[1], false, false);
        acc[2] = __builtin_amdgcn_wmma_f32_16x16x4_f32(false, a, false, b2,
                                                       (short)0, acc[2], false, false);
        acc[3] = __builtin_amdgcn_wmma_f32_16x16x4_f32(false, a, false, b3,
                                                       (short)0, acc[3], false, false);
    }

    // Epilogue: (acc + bias) * inv_temp ; C/D layout: VGPR j -> M = j (+8 for
    // lanes 16-31), N = lane&15 within each 16-wide tile.
    float invt = ws[64];
    float bia0 = ws[0 * 16 + m];
    float bia1 = ws[1 * 16 + m];
    float bia2 = ws[2 * 16 + m];
    float bia3 = ws[3 * 16 + m];
    int mrow = (lane >> 4) * 8;
    #pragma unroll
    for (int j = 0; j < 8; ++j) {
        size_t r = (size_t)(row_base + mrow + j) * RT_E;
        logits[r + 0 * 16 + m] = (acc[0][j] + bia0) * invt;
        logits[r + 1 * 16 + m] = (acc[1][j] + bia1) * invt;
        logits[r + 2 * 16 + m] = (acc[2][j] + bia2) * invt;
        logits[r + 3 * 16 + m] = (acc[3][j] + bia3) * invt;
    }
}

// ---------------------------------------------------------------------------
// Kernel 3: per-row top-2, softmax weights, full-softmax stats.
// One wave (32 lanes) per row; lane covers experts {lane, lane+32}.
// ---------------------------------------------------------------------------
__global__ __launch_bounds__(256) void router_topk_kernel(
    const float* __restrict__ logits,
    float* __restrict__ out_idx,
    float* __restrict__ out_wts,
    float* __restrict__ wsacc) {           // ws+65: [0..63] load, [64] entropy
    __shared__ float lds_load[64];
    __shared__ float lds_ent;
    const int lane = threadIdx.x & 31;
    const int wave = threadIdx.x >> 5;

    if (threadIdx.x < 64) lds_load[threadIdx.x] = 0.f;
    if (threadIdx.x == 0) lds_ent = 0.f;
    __syncthreads();

    for (int r = 0; r < 8; ++r) {
        int row = blockIdx.x * 64 + wave * 8 + r;
        float v0 = logits[(size_t)row * RT_E + lane];
        float v1 = logits[(size_t)row * RT_E + 32 + lane];

        // local sorted top-2 (lower index wins ties: lane < lane+32)
        bool f = (v0 >= v1);
        float a1 = f ? v0 : v1;  int i1 = f ? lane : lane + 32;
        float a2 = f ? v1 : v0;  int i2 = f ? lane + 32 : lane;

        // wave32 butterfly top-2 merge
        #pragma unroll
        for (int off = 16; off >= 1; off >>= 1) {
            float b1 = __shfl_xor(a1, off, 32); int j1 = __shfl_xor(i1, off, 32);
            float b2 = __shfl_xor(a2, off, 32); int j2 = __shfl_xor(i2, off, 32);
            bool t  = (a1 > b1) || (a1 == b1 && i1 < j1);
            float n1  = t ? a1 : b1;  int ni1 = t ? i1 : j1;
            float c2v = t ? a2 : b2;  int c2i = t ? i2 : j2;  // winner's 2nd
            float c1v = t ? b1 : a1;  int c1i = t ? j1 : i1;  // loser's 1st
            bool t2 = (c2v > c1v) || (c2v == c1v && c2i < c1i);
            a1 = n1; i1 = ni1;
            a2 = t2 ? c2v : c1v; i2 = t2 ? c2i : c1i;
        }

        // full softmax stats (row max == a1)
        float e0 = __expf(v0 - a1), e1 = __expf(v1 - a1);
        float s = e0 + e1;
        #pragma unroll
        for (int off = 16; off >= 1; off >>= 1) s += __shfl_xor(s, off, 32);
        float p0 = e0 / s, p1 = e1 / s;
        float ent = -(p0 * __logf(p0 + RT_EPS) + p1 * __logf(p1 + RT_EPS));
        #pragma unroll
        for (int off = 16; off >= 1; off >>= 1) ent += __shfl_xor(ent, off, 32);

        atomicAdd(&lds_load[lane], p0);        // ds_add_f32
        atomicAdd(&lds_load[lane + 32], p1);

        if (lane == 0) {
            atomicAdd(&lds_ent, ent);
            float w2 = __expf(a2 - a1);        // a2 <= a1
            float denom = 1.0f + w2;
            out_idx[(size_t)row * 2 + 0] = (float)i1;
            out_idx[(size_t)row * 2 + 1] = (float)i2;
            out_wts[(size_t)row * 2 + 0] = 1.0f / denom;
            out_wts[(size_t)row * 2 + 1] = w2 / denom;
        }
    }
    __syncthreads();
    if (threadIdx.x < 64) atomicAdd(&wsacc[threadIdx.x], lds_load[threadIdx.x]);
    if (threadIdx.x == 0) atomicAdd(&wsacc[64], lds_ent);
}

// ---------------------------------------------------------------------------
// Kernel 4: finalize expert_load mean, unbiased variance, entropy mean.
// ---------------------------------------------------------------------------
__global__ void router_finalize_kernel(const float* __restrict__ wsacc,
                                       float* __restrict__ out_load,
                                       float* __restrict__ out_var,
                                       float* __restrict__ out_ent) {
    __shared__ float sm[64];
    int t = threadIdx.x;                 // 64 threads
    float load = wsacc[t] * (1.0f / (float)RT_N);
    out_load[t] = load;
    sm[t] = load;
    __syncthreads();
    float mean = 0.f;
    for (int i = 0; i < 64; ++i) mean += sm[i];
    mean *= (1.0f / 64.0f);
    float d = load - mean;
    __syncthreads();
    sm[t] = d * d;
    __syncthreads();
    if (t == 0) {
        float var = 0.f;
        for (int i = 0; i < 64; ++i) var += sm[i];
        out_var[0] = var / 63.0f;                       // ddof=1
        out_ent[0] = wsacc[64] * (1.0f / (float)RT_N);
    }
}

// ---------------------------------------------------------------------------
extern "C" void kernel_launch(void* const* d_in, const int* in_sizes, int n_in,
                              void* d_out, int out_size, void* d_ws, size_t ws_size,
                              hipStream_t stream) {
    const float* hs   = (const float*)d_in[0];  // [16384, 4096]
    const float* w    = (const float*)d_in[1];  // [64, 4096]
    const float* imp  = (const float*)d_in[2];  // [64]
    const float* temp = (const float*)d_in[3];  // [1]
    (void)in_sizes; (void)n_in; (void)out_size; (void)ws_size;

    float* wsf   = (float*)d_ws;
    float* bpack = wsf + 256;            // 1 MB packed B fragments

    float* out        = (float*)d_out;
    float* out_logits = out;                              // [N, 64]
    float* out_idx    = out_logits + (size_t)RT_N * RT_E; // [N, 2] (as float)
    float* out_wts    = out_idx + (size_t)RT_N * 2;       // [N, 2]
    float* out_load   = out_wts + (size_t)RT_N * 2;       // [64]
    float* out_var    = out_load + RT_E;                  // [1]
    float* out_ent    = out_var + 1;                      // [1]

    router_prep_kernel<<<1, 64, 0, stream>>>(imp, temp, wsf);
    router_pack_kernel<<<512, 256, 0, stream>>>(w, bpack);
    router_gemm_kernel<<<RT_N / 128, 256, 0, stream>>>(hs, bpack, wsf, out_logits);
    router_topk_kernel<<<RT_N / 64, 256, 0, stream>>>(out_logits, out_idx, out_wts,
                                                      wsf + 65);
    router_finalize_kernel<<<1, 64, 0, stream>>>(wsf + 65, out_load, out_var, out_ent);
}